// IE_66640712564827
// MI455X (gfx1250) — compile-verified
//
#include <hip/hip_runtime.h>
#include <math.h>

typedef __attribute__((ext_vector_type(2))) float v2f;
typedef __attribute__((ext_vector_type(8))) float v8f;

#define NFEAT 12
#define NMASK 4095   // 2^12 - 1  (columns of fl/fu, rows of fmat)
#define NV    4094   // 2^12 - 2  (length of ie_vars / fm scan)
#define NCOL  8190   // 2*(2^12-1) (columns of fmat)
#define NBATCH 64

// d_ws float layout
#define WS_FM2 0      // 8190 floats: FM2[2k]=+FM[k], FM2[2k+1]=-FM[k]
#define WS_W   8192   // 4095 floats: w = fmat @ FM2

#if defined(__has_builtin)
#if __has_builtin(__builtin_amdgcn_global_load_async_to_lds_b32)
#define HAVE_ASYNC_LDS 1
#endif
#endif

// ---------------------------------------------------------------------------
// Kernel 1: build FM via popcount-level DP (replaces the sequential scan),
// then emit the signed/interleaved FM2 vector.
// Masks of popcount L depend only on popcount L-1 -> 11 barrier levels.
// ---------------------------------------------------------------------------
__global__ void build_fm2_kernel(const float* __restrict__ ie,
                                 float* __restrict__ ws) {
  __shared__ float fms[NMASK + 1];
  const int tid = threadIdx.x;

  for (int lvl = 1; lvl <= 11; ++lvl) {   // masks <= 4094 have popcount <= 11
    for (int m = tid + 1; m <= NV; m += blockDim.x) {
      if (__popc(m) == lvl) {
        float mx = 0.0f;
        if (lvl > 1) {
          mx = -1e30f;
#pragma unroll
          for (int b = 0; b < NFEAT; ++b)
            if (m & (1 << b)) mx = fmaxf(mx, fms[m ^ (1 << b)]);
        }
        fms[m] = fabsf(ie[m - 1]) + mx;
      }
    }
    __syncthreads();
  }

  // FM[k] = min(fm[k], 1) for k<NV ; FM[NV] = 1  (appended one)
  for (int c = tid; c < NCOL; c += blockDim.x) {
    const int k = c >> 1;  // 0..4094
    const float fmv = (k < NV) ? fminf(fms[k + 1], 1.0f) : 1.0f;
    ws[WS_FM2 + c] = (c & 1) ? -fmv : fmv;
  }
}

// ---------------------------------------------------------------------------
// Kernel 2: w[row] = dot(fmat[row, :], FM2)   (the 134 MB streaming part)
// One wave32 per row; FM2 staged in LDS via the gfx1250 async-to-LDS path
// when available (ASYNCcnt-tracked), else plain loads.  float4 coalesced
// streaming: 32 lanes x 16 B = 512 B per wave-iteration.
// ---------------------------------------------------------------------------
__global__ void matvec_w_kernel(const float* __restrict__ fmat,
                                float* __restrict__ ws) {
  __shared__ float fm2[NCOL];
  const int tid = threadIdx.x;

#ifdef HAVE_ASYNC_LDS
  {
    typedef __attribute__((address_space(1))) int* gptr_t;
    typedef __attribute__((address_space(3))) int* lptr_t;
    gptr_t gsrc = (gptr_t)(ws + WS_FM2);
    lptr_t ldst = (lptr_t)fm2;
    for (int c = tid; c < NCOL; c += blockDim.x)
      __builtin_amdgcn_global_load_async_to_lds_b32(gsrc + c, ldst + c, 0, 0);
#if __has_builtin(__builtin_amdgcn_s_wait_asynccnt)
    __builtin_amdgcn_s_wait_asynccnt(0);
#else
    asm volatile("s_wait_asynccnt 0" ::: "memory");
#endif
  }
#else
  for (int c = tid; c < NCOL; c += blockDim.x) fm2[c] = ws[WS_FM2 + c];
#endif
  __syncthreads();

  const int wave = tid >> 5;
  const int lane = tid & 31;
  const int row = blockIdx.x * 8 + wave;
  if (row >= NMASK) return;

  const float* rp = fmat + (size_t)row * NCOL;
  float acc = 0.0f;
#pragma unroll 4
  for (int it = 0; it < 64; ++it) {       // 64*128 = 8192 >= 8190 cols
    const int col = it * 128 + lane * 4;
    if (col + 2048 < NCOL) __builtin_prefetch(rp + col + 2048, 0, 1);
    if (col + 3 < NCOL) {
      const float4 fv = *(const float4*)(rp + col);
      acc += fv.x * fm2[col] + fv.y * fm2[col + 1] +
             fv.z * fm2[col + 2] + fv.w * fm2[col + 3];
    } else {
#pragma unroll
      for (int i = 0; i < 4; ++i)
        if (col + i < NCOL) acc += rp[col + i] * fm2[col + i];
    }
  }
  // wave32 reduction
#pragma unroll
  for (int off = 16; off > 0; off >>= 1) acc += __shfl_xor(acc, off, 32);
  if (lane == 0) ws[WS_W + row] = acc;
}

// Branch-free on-the-fly subset product: prod_{b in (k+1)} xv[b].
// For the padded slot k = NMASK (mask 4096) no low bit is set -> returns 1.0,
// which is harmless because its B-side weight is the wsh[NMASK]=0 pad.
__device__ __forceinline__ float subset_prod(const float xv[NFEAT], int k) {
  const int mask = k + 1;
  float p = 1.0f;
#pragma unroll
  for (int b = 0; b < NFEAT; ++b)
    p *= (mask & (1 << b)) ? xv[b] : 1.0f;
  return p;
}

// ---------------------------------------------------------------------------
// Kernel 3: y = [fl; fu] (128 x 4095) @ w via V_WMMA_F32_16X16X4_F32.
// Wave i owns A-tile rows 16i..16i+15 (A 16x4 f32: lanes 0-15 hold K=0,1 and
// lanes 16-31 hold K=2,3 of the same 16 rows). w sits in column N=0 of B
// (only lanes 0 and 16 carry nonzero B), so D column 0 (lanes 0/16, 8 VGPRs
// each) is the result.  left = min(y_l, y_u), right = y_u.
// Loop body is fully branch-free: EXEC stays all-1s around every WMMA.
// ---------------------------------------------------------------------------
__global__ void final_wmma_kernel(const float* __restrict__ x,
                                  const float* __restrict__ ws,
                                  float* __restrict__ out) {
  __shared__ float wsh[NMASK + 1];   // +1 zero pad for the K tail
  __shared__ float ysh[2 * NBATCH];
  const int tid = threadIdx.x;
  for (int c = tid; c < NMASK; c += blockDim.x) wsh[c] = ws[WS_W + c];
  if (tid == 0) wsh[NMASK] = 0.0f;
  __syncthreads();

  const int wave = tid >> 5;               // 0..7  -> 128 F-rows
  const int lane = tid & 31;
  const int rowIdx = wave * 16 + (lane & 15);      // 0..127
  const int batch = rowIdx & (NBATCH - 1);
  const int xoff = (rowIdx < NBATCH) ? 0 : NFEAT;  // fl uses xl, fu uses xu

  float xv[NFEAT];
#pragma unroll
  for (int b = 0; b < NFEAT; ++b) xv[b] = x[batch * 2 * NFEAT + xoff + b];

  const int khalf = (lane >> 4) << 1;      // lanes 0-15 -> K=0,1 ; 16-31 -> K=2,3
  const float sel = ((lane & 15) == 0) ? 1.0f : 0.0f;  // only lanes 0,16 feed B

  v8f acc = {};
#pragma unroll 4
  for (int k0 = 0; k0 < NMASK; k0 += 4) {  // 1024 WMMA accumulations
    const int kk = k0 + khalf;
    v2f a;
    a[0] = subset_prod(xv, kk);
    a[1] = subset_prod(xv, kk + 1);
    v2f bb;
    bb[0] = sel * wsh[kk];                 // lane 0: w[k0]   ; lane 16: w[k0+2]
    bb[1] = sel * wsh[kk + 1];             // lane 0: w[k0+1] ; lane 16: w[k0+3]
    acc = __builtin_amdgcn_wmma_f32_16x16x4_f32(
        /*neg_a=*/false, a, /*neg_b=*/false, bb,
        /*c_mod=*/(short)0, acc, /*reuse_a=*/false, /*reuse_b=*/false);
  }

  // D column N=0: lane 0 holds M=0..7 in VGPRs 0..7, lane 16 holds M=8..15
  const int m0 = wave * 16;
  if (lane == 0) {
#pragma unroll
    for (int r = 0; r < 8; ++r) ysh[m0 + r] = acc[r];
  }
  if (lane == 16) {
#pragma unroll
    for (int r = 0; r < 8; ++r) ysh[m0 + 8 + r] = acc[r];
  }
  __syncthreads();

  if (tid < NBATCH) {
    const float yl = ysh[tid];
    const float yu = ysh[NBATCH + tid];
    out[tid] = fminf(yl, yu);            // left
    out[NBATCH + tid] = yu;              // right
  }
}

// ---------------------------------------------------------------------------
extern "C" void kernel_launch(void* const* d_in, const int* in_sizes, int n_in,
                              void* d_out, int out_size, void* d_ws, size_t ws_size,
                              hipStream_t stream) {
  (void)in_sizes; (void)n_in; (void)out_size; (void)ws_size;
  const float* x    = (const float*)d_in[0];   // (64, 24)
  const float* ie   = (const float*)d_in[1];   // (4094, 1)
  const float* fmat = (const float*)d_in[2];   // (4095, 8190)
  float* out = (float*)d_out;                  // 128 floats
  float* ws  = (float*)d_ws;

  build_fm2_kernel<<<1, 256, 0, stream>>>(ie, ws);
  matvec_w_kernel<<<512, 256, 0, stream>>>(fmat, ws);   // 512*8 waves >= 4095 rows
  final_wmma_kernel<<<1, 256, 0, stream>>>(x, ws, out);
}